// DeepseekV4HyperConnection_13761075216602
// MI455X (gfx1250) — compile-verified
//
#include <hip/hip_runtime.h>

typedef __attribute__((ext_vector_type(2))) float v2f;
typedef __attribute__((ext_vector_type(8))) float v8f;

#define K_DIM 8192      // H*D
#define D_DIM 2048
#define H_DIM 4
#define TOK   8         // tokens per block (8*8200*4B = 256.25 KB LDS)
#define TPB   512       // 16 waves
#define WAVES 16
#define KPW   (K_DIM / WAVES)   // 512 K per wave
#define XSTR  8200      // padded token stride in LDS (bank-conflict-free A reads)

__device__ __forceinline__ float sigmoidf(float x) { return 1.0f / (1.0f + __expf(-x)); }

// ---------------------------------------------------------------------------
// Kernel 0: swizzle fn [24 x 8192] into per-lane WMMA B-fragment order in ws.
// Fragment f = k4*2 + tile (k4 = K/4 position, tile = N-tile 0/1), 32 lanes of
// float2 each: lane L holds { fn[n][k], fn[n][k+1] } with n = tile*16 + (L&15)
// (clamped to 23; padded columns are never read back) and k = k4*4 + (L>>4)*2.
// ---------------------------------------------------------------------------
__global__ void fn_swizzle_kernel(const float* __restrict__ fn, v2f* __restrict__ wsB) {
  int g    = blockIdx.x * blockDim.x + threadIdx.x;   // 0 .. 131071
  int lane = g & 31;
  int f    = g >> 5;                                  // 0..4095
  int tile = f & 1;
  int k4   = f >> 1;                                  // 0..2047
  int n    = tile * 16 + (lane & 15);
  if (n > 23) n = 23;
  int kk   = k4 * 4 + ((lane >> 4) << 1);
  const float* p = fn + (long)n * K_DIM + kk;
  v2f v; v.x = p[0]; v.y = p[1];
  wsB[(long)f * 32 + lane] = v;
}

// ---------------------------------------------------------------------------
// Fused kernel: one block = 8 tokens.
//  Phase 1: global -> LDS copy of raw streams (coalesced float4) + sum-of-squares
//  Phase 2: WMMA f32 16x16x4 projection (raw dots), cross-wave reduce, fold RMS
//  Phase 3: gates + per-token 4x4 Sinkhorn (one thread per token)
//  Phase 4: weighted collapse straight from LDS (raw data), coalesced stores
// ---------------------------------------------------------------------------
__global__ void __launch_bounds__(TPB, 1)
hyperconn_fused_kernel(const float* __restrict__ hs, const float* __restrict__ fn,
                       const float* __restrict__ base, const float* __restrict__ scale,
                       float* __restrict__ out, const v2f* __restrict__ wsB,
                       int use_ws, int ntok)
{
  extern __shared__ float smem[];
  float* xs   = smem;                        // TOK * XSTR      raw token data
  float* cred = smem + TOK * XSTR;           // WAVES * 512     partial C tiles
  float* mixs = cred + WAVES * 512;          // 512             reduced mix (raw)
  float* ssq  = mixs + 512;                  // 8
  float* rmsv = ssq + 8;                     // 8
  float* preL = rmsv + 8;                    // 32

  const int  tid  = threadIdx.x;
  const int  lane = tid & 31;
  const int  w    = tid >> 5;                // wave id 0..15
  const long tok0 = (long)blockIdx.x * TOK;

  if (tid < TOK) ssq[tid] = 0.0f;

  // ---- Phase 1: stage 8 tokens (raw) into LDS, accumulate sum of squares ----
  const float4* src = (const float4*)(hs + tok0 * K_DIM);
  float partial[TOK];
  #pragma unroll
  for (int t = 0; t < TOK; ++t) partial[t] = 0.0f;
  #pragma unroll
  for (int it = 0; it < 32; ++it) {          // 32 * 512 = 16384 float4 = 8 tokens
    int   idx = it * TPB + tid;
    int   t   = it >> 2;                     // compile-time token id
    int   k4  = idx & 2047;
    float4 v  = src[idx];
    partial[t] += v.x * v.x + v.y * v.y + v.z * v.z + v.w * v.w;
    *(float4*)(xs + t * XSTR + k4 * 4) = v;
  }
  __syncthreads();                           // ssq zeros + xs visible
  #pragma unroll
  for (int t = 0; t < TOK; ++t) {
    float v = partial[t];
    #pragma unroll
    for (int off = 16; off; off >>= 1) v += __shfl_xor(v, off, 32);
    if (lane == 0) atomicAdd(&ssq[t], v);
  }
  __syncthreads();
  if (tid < TOK) rmsv[tid] = rsqrtf(ssq[tid] * (1.0f / K_DIM) + 1e-6f);

  // ---- Phase 2: WMMA projection. mix_raw[t][n] = sum_k x[t][k]*fn[n][k] ----
  // A 16x4 f32 fragment: lanes 0-15 rows (tokens, mirrored 8..15), K=0,1;
  // lanes 16-31 K=2,3. One ds b64 per lane per step; 32 distinct banks.
  const float* aptr = xs + (lane & 7) * XSTR + ((lane >> 4) << 1);

  const v2f* b0; const v2f* b1; int bstride;
  if (use_ws) {
    long f0 = ((long)w * (KPW / 4)) * 2;     // starting fragment for this wave
    b0 = wsB + (f0 + 0) * 32 + lane;
    b1 = wsB + (f0 + 1) * 32 + lane;
    bstride = 64;                            // 2 fragments per k4 step
  } else {                                   // fallback: gather straight from fn
    int dlt = (lane >> 4) << 1;
    int n0  = lane & 15;
    int n1  = 16 + (lane & 15); if (n1 > 23) n1 = 23;
    b0 = (const v2f*)(fn + (long)n0 * K_DIM + dlt) + (long)(w * (KPW / 4)) * 2;
    b1 = (const v2f*)(fn + (long)n1 * K_DIM + dlt) + (long)(w * (KPW / 4)) * 2;
    bstride = 2;
  }

  v8f c0 = {}; v8f c1 = {};
  const int kbase = w * KPW;
  for (int k4 = 0; k4 < KPW / 4; ++k4) {
    int k0  = kbase + k4 * 4;
    v2f a   = *(const v2f*)(aptr + k0);
    v2f bb0 = *b0; b0 += bstride;
    v2f bb1 = *b1; b1 += bstride;
    // 8 args: (neg_a, A, neg_b, B, c_mod, C, reuse_a, reuse_b)
    c0 = __builtin_amdgcn_wmma_f32_16x16x4_f32(false, a, false, bb0, (short)0, c0, false, false);
    c1 = __builtin_amdgcn_wmma_f32_16x16x4_f32(false, a, false, bb1, (short)0, c1, false, false);
  }
  #pragma unroll
  for (int g = 0; g < 8; ++g) {              // store partial C tiles
    cred[w * 512 +       g * 32 + lane] = c0[g];
    cred[w * 512 + 256 + g * 32 + lane] = c1[g];
  }
  __syncthreads();
  {                                          // cross-wave reduce: slot = tid
    float s = 0.0f;
    #pragma unroll
    for (int ww = 0; ww < WAVES; ++ww) s += cred[ww * 512 + tid];
    mixs[tid] = s;                           // [tile*256 + row*32 + col]
  }
  __syncthreads();

  // ---- Phase 3: gates + Sinkhorn, one thread per token ----
  float* out_post = out;
  float* out_comb = out + (long)ntok * 4;
  float* out_col  = out + (long)ntok * 20;
  if (tid < TOK) {
    const int  t  = tid;
    const float r = rmsv[t];                 // RMS folds into mix (linearity)
    const long tk = tok0 + t;
    const float s0 = scale[0], s1 = scale[1], s2 = scale[2];
    #pragma unroll
    for (int h = 0; h < 4; ++h) {
      float mpre  = mixs[t * 32 + h]     * r;
      float mpost = mixs[t * 32 + 4 + h] * r;
      preL[t * 4 + h]      = sigmoidf(mpre  * s0 + base[h])     + 1e-5f;
      out_post[tk * 4 + h] = sigmoidf(mpost * s1 + base[4 + h]) + 1e-5f;
    }
    float cmb[4][4];
    #pragma unroll
    for (int i = 0; i < 4; ++i)
      #pragma unroll
      for (int j = 0; j < 4; ++j) {
        int   m = 8 + i * 4 + j;
        float v = (m < 16 ? mixs[t * 32 + m] : mixs[256 + t * 32 + (m - 16)]) * r;
        cmb[i][j] = sigmoidf(v * s2 + base[m]) + 1e-5f;
      }
    #pragma unroll
    for (int it = 0; it < 10; ++it) {
      #pragma unroll
      for (int i = 0; i < 4; ++i) {          // row normalize
        float inv = 1.0f / (cmb[i][0] + cmb[i][1] + cmb[i][2] + cmb[i][3] + 1e-5f);
        cmb[i][0] *= inv; cmb[i][1] *= inv; cmb[i][2] *= inv; cmb[i][3] *= inv;
      }
      #pragma unroll
      for (int j = 0; j < 4; ++j) {          // column normalize
        float inv = 1.0f / (cmb[0][j] + cmb[1][j] + cmb[2][j] + cmb[3][j] + 1e-5f);
        cmb[0][j] *= inv; cmb[1][j] *= inv; cmb[2][j] *= inv; cmb[3][j] *= inv;
      }
    }
    float* oc = out_comb + tk * 16;
    #pragma unroll
    for (int i = 0; i < 4; ++i)
      #pragma unroll
      for (int j = 0; j < 4; ++j) oc[i * 4 + j] = cmb[i][j];
  }
  __syncthreads();

  // ---- Phase 4: collapse = sum_h pre[h] * raw_x[h][:], from LDS ----
  float* ocol = out_col + tok0 * D_DIM;
  #pragma unroll
  for (int t = 0; t < TOK; ++t) {
    float p0 = preL[t * 4 + 0], p1 = preL[t * 4 + 1];
    float p2 = preL[t * 4 + 2], p3 = preL[t * 4 + 3];
    const float* xb = xs + t * XSTR;
    float4 v0 = *(const float4*)(xb + 0 * D_DIM + tid * 4);
    float4 v1 = *(const float4*)(xb + 1 * D_DIM + tid * 4);
    float4 v2 = *(const float4*)(xb + 2 * D_DIM + tid * 4);
    float4 v3 = *(const float4*)(xb + 3 * D_DIM + tid * 4);
    float4 o;
    o.x = p0 * v0.x + p1 * v1.x + p2 * v2.x + p3 * v3.x;
    o.y = p0 * v0.y + p1 * v1.y + p2 * v2.y + p3 * v3.y;
    o.z = p0 * v0.z + p1 * v1.z + p2 * v2.z + p3 * v3.z;
    o.w = p0 * v0.w + p1 * v1.w + p2 * v2.w + p3 * v3.w;
    *(float4*)(ocol + t * D_DIM + tid * 4) = o;
  }
}

extern "C" void kernel_launch(void* const* d_in, const int* in_sizes, int n_in,
                              void* d_out, int out_size, void* d_ws, size_t ws_size,
                              hipStream_t stream) {
  const float* hs    = (const float*)d_in[0];
  const float* fn    = (const float*)d_in[1];
  const float* base  = (const float*)d_in[2];
  const float* scale = (const float*)d_in[3];
  float* out = (float*)d_out;

  const int ntok   = in_sizes[0] / K_DIM;    // 8192
  const int blocks = ntok / TOK;             // 1024

  const size_t wsNeed = (size_t)2048 * 2 * 32 * sizeof(v2f);   // 1 MiB
  const int use_ws = (ws_size >= wsNeed) ? 1 : 0;
  v2f* wsB = (v2f*)d_ws;
  if (use_ws) fn_swizzle_kernel<<<512, 256, 0, stream>>>(fn, wsB);

  const size_t lds = (size_t)(TOK * XSTR + WAVES * 512 + 512 + 48) * sizeof(float); // ~297 KB
  hipFuncSetAttribute(reinterpret_cast<const void*>(hyperconn_fused_kernel),
                      hipFuncAttributeMaxDynamicSharedMemorySize, (int)lds);
  hyperconn_fused_kernel<<<blocks, TPB, lds, stream>>>(hs, fn, base, scale, out,
                                                       wsB, use_ws, ntok);
}